// MyHeteroConv_59854664237663
// MI455X (gfx1250) — compile-verified
//
#include <hip/hip_runtime.h>
#include <hip/hip_bf16.h>

typedef float v2f __attribute__((ext_vector_type(2)));
typedef float v8f __attribute__((ext_vector_type(8)));

#define DIMF   128
#define NUSER  100000
#define NPOI   50000
#define WLSTRIDE 132   // 128 + 4 pad: (4n + c) % 64 covers all banks for float2 loads

// ---------------------------------------------------------------------------
// Y[M,128] = X[M,128] * W[128,128]  (f32, V_WMMA_F32_16X16X4_F32)
// 256 threads = 8 waves; each wave computes a 16-row x 128-col stripe.
// ---------------------------------------------------------------------------
__global__ __launch_bounds__(256) void gemm128_wmma_f32(
    const float* __restrict__ X, const float* __restrict__ W,
    float* __restrict__ Y, int M) {
  __shared__ float Wl[DIMF * WLSTRIDE];  // W transposed: Wl[n*132 + k] = W[k,n]

  for (int idx = threadIdx.x; idx < DIMF * DIMF; idx += 256) {
    int k = idx >> 7;
    int n = idx & 127;
    Wl[n * WLSTRIDE + k] = W[idx];
  }
  __syncthreads();

  const int wave = threadIdx.x >> 5;
  const int lane = threadIdx.x & 31;
  const int half = lane >> 4;   // 0: lanes 0-15, 1: lanes 16-31
  const int lm   = lane & 15;
  const int row0 = (blockIdx.x * 8 + wave) * 16;
  if (row0 >= M) return;        // wave-uniform: EXEC stays all-ones for WMMA

  v8f acc[8] = {};

  // A fragment (16x4 f32): lane row = row0 + lm, VGPR v holds K = k0 + 2*half + v
  const float* xrow = X + (size_t)(row0 + lm) * DIMF + 2 * half;

  #pragma unroll 4
  for (int k0 = 0; k0 < DIMF; k0 += 4) {
    v2f a = *(const v2f*)(xrow + k0);
    #pragma unroll
    for (int j = 0; j < 8; ++j) {
      // B fragment (4x16 f32): lane col = j*16 + lm, VGPR v holds K = k0 + 2*half + v
      v2f b = *(const v2f*)(Wl + (j * 16 + lm) * WLSTRIDE + k0 + 2 * half);
      acc[j] = __builtin_amdgcn_wmma_f32_16x16x4_f32(
          false, a, false, b, (short)0, acc[j], false, false);
    }
  }

  // C/D layout: VGPR r -> row = row0 + r + 8*half, col = j*16 + lm
  #pragma unroll
  for (int j = 0; j < 8; ++j) {
    #pragma unroll
    for (int r = 0; r < 8; ++r) {
      Y[(size_t)(row0 + r + 8 * half) * DIMF + j * 16 + lm] = acc[j][r];
    }
  }
}

// ---------------------------------------------------------------------------
__global__ __launch_bounds__(256) void zero_i32(int* __restrict__ p, int n) {
  int i = blockIdx.x * 256 + threadIdx.x;
  if (i < n) p[i] = 0;
}

__global__ __launch_bounds__(256) void count_edges(
    const int* __restrict__ etgt, int* __restrict__ cnt, int E) {
  int i = blockIdx.x * 256 + threadIdx.x;
  if (i < E) atomicAdd(cnt + etgt[i], 1);
}

// One wave per edge: 32 lanes x float4 = 128 floats. msg scaled by 1/max(cnt,1)
// and atomically accumulated into out (which already holds the tgt GEMM term).
// unsafeAtomicAdd guarantees the native global_atomic_add_f32 path (no CAS loop);
// this phase is the roofline bottleneck (~2 GB gather + 256M dword atomics in L2).
__global__ __launch_bounds__(256) void scatter_mean_add(
    const float* __restrict__ src, const int* __restrict__ esrc,
    const int* __restrict__ etgt, const int* __restrict__ cnt,
    float* __restrict__ out, int E) {
  int edge = blockIdx.x * 8 + (threadIdx.x >> 5);
  if (edge >= E) return;
  int lane = threadIdx.x & 31;
  int s = esrc[edge];
  int t = etgt[edge];
  int c = cnt[t];
  float scale = 1.0f / (float)(c > 0 ? c : 1);
  float4 v = *(const float4*)(src + (size_t)s * DIMF + lane * 4);
  float* op = out + (size_t)t * DIMF + lane * 4;
  unsafeAtomicAdd(op + 0, v.x * scale);
  unsafeAtomicAdd(op + 1, v.y * scale);
  unsafeAtomicAdd(op + 2, v.z * scale);
  unsafeAtomicAdd(op + 3, v.w * scale);
}

__global__ __launch_bounds__(256) void relu_f32x4(float* __restrict__ p, int n4) {
  int i = blockIdx.x * 256 + threadIdx.x;
  if (i < n4) {
    float4 v = ((float4*)p)[i];
    v.x = fmaxf(v.x, 0.0f);
    v.y = fmaxf(v.y, 0.0f);
    v.z = fmaxf(v.z, 0.0f);
    v.w = fmaxf(v.w, 0.0f);
    ((float4*)p)[i] = v;
  }
}

// ---------------------------------------------------------------------------
extern "C" void kernel_launch(void* const* d_in, const int* in_sizes, int n_in,
                              void* d_out, int out_size, void* d_ws, size_t ws_size,
                              hipStream_t stream) {
  const float* x_user    = (const float*)d_in[0];
  const float* x_poi     = (const float*)d_in[1];
  const float* W_u2p_src = (const float*)d_in[2];
  const float* W_u2p_tgt = (const float*)d_in[3];
  const float* W_p2u_src = (const float*)d_in[4];
  const float* W_p2u_tgt = (const float*)d_in[5];
  const int* e_u2p_src   = (const int*)d_in[6];
  const int* e_u2p_tgt   = (const int*)d_in[7];
  const int* e_p2u_src   = (const int*)d_in[8];
  const int* e_p2u_tgt   = (const int*)d_in[9];
  const int E1 = in_sizes[6];
  const int E2 = in_sizes[8];

  float* out_user = (float*)d_out;                          // [NUSER,128]
  float* out_poi  = (float*)d_out + (size_t)NUSER * DIMF;   // [NPOI,128]

  char* ws = (char*)d_ws;
  float* src  = (float*)ws;  ws += (size_t)NUSER * DIMF * sizeof(float);
  float* src2 = (float*)ws;  ws += (size_t)NPOI  * DIMF * sizeof(float);
  int* cnt_poi  = (int*)ws;  ws += (size_t)NPOI  * sizeof(int);
  int* cnt_user = (int*)ws;  // cnt_poi and cnt_user contiguous

  // 1) zero + count edge targets (cnt_poi/cnt_user contiguous -> one zero pass)
  {
    int n = NPOI + NUSER;
    zero_i32<<<(n + 255) / 256, 256, 0, stream>>>(cnt_poi, n);
  }
  count_edges<<<(E1 + 255) / 256, 256, 0, stream>>>(e_u2p_tgt, cnt_poi, E1);
  count_edges<<<(E2 + 255) / 256, 256, 0, stream>>>(e_p2u_tgt, cnt_user, E2);

  // 2) the four GEMMs (tgt terms go straight into d_out)
  auto gblocks = [](int M) { return ((M / 16) + 7) / 8; };
  gemm128_wmma_f32<<<gblocks(NUSER), 256, 0, stream>>>(x_user, W_u2p_src, src,      NUSER);
  gemm128_wmma_f32<<<gblocks(NPOI),  256, 0, stream>>>(x_poi,  W_u2p_tgt, out_poi,  NPOI);
  gemm128_wmma_f32<<<gblocks(NPOI),  256, 0, stream>>>(x_poi,  W_p2u_src, src2,     NPOI);
  gemm128_wmma_f32<<<gblocks(NUSER), 256, 0, stream>>>(x_user, W_p2u_tgt, out_user, NUSER);

  // 3) gather + mean-scaled atomic scatter into out (adds to tgt term)
  scatter_mean_add<<<(E1 + 7) / 8, 256, 0, stream>>>(src,  e_u2p_src, e_u2p_tgt,
                                                     cnt_poi,  out_poi,  E1);
  scatter_mean_add<<<(E2 + 7) / 8, 256, 0, stream>>>(src2, e_p2u_src, e_p2u_tgt,
                                                     cnt_user, out_user, E2);

  // 4) ReLU over the whole output
  {
    int n4 = (NUSER + NPOI) * DIMF / 4;
    relu_f32x4<<<(n4 + 255) / 256, 256, 0, stream>>>((float*)d_out, n4);
  }
}